// CenterHead_74105365725359
// MI455X (gfx1250) — compile-verified
//
#include <hip/hip_runtime.h>
#include <math.h>

typedef __attribute__((ext_vector_type(2))) float v2f;
typedef __attribute__((ext_vector_type(8))) float v8f;

#define BATCH 16
#define CIN   256
#define HH    128
#define WW    128
#define HWSZ  (HH * WW)      // 16384
#define KSEL  300

// ---------------------------------------------------------------------------
// Kernel 1: fused 1x1 convs via V_WMMA_F32_16X16X4_F32 + sigmoid + score.
// A = padded weights (16 outputs x K), rows 3..15 zero, staged in LDS.
// B = x tile (K=4 x 16 pixels). D(16x16): lanes 0..15 own one pixel each,
// accumulator regs 0/1/2 = reg0 / reg1 / conf-logit.
// Each wave: 2 pixel tiles (32 px), block = 8 waves = 256 px, grid = 1024.
// ---------------------------------------------------------------------------
__global__ __launch_bounds__(256) void head_gemm_kernel(
    const float* __restrict__ x, const float* __restrict__ w_reg,
    const float* __restrict__ b_reg, const float* __restrict__ w_conf,
    const float* __restrict__ b_conf, float4* __restrict__ payload,
    float* __restrict__ scores)
{
    __shared__ float wpad[16 * CIN];   // 16 KB: row0=w_reg0, row1=w_reg1, row2=w_conf, rest 0

    const int tid = threadIdx.x;
    for (int i = tid; i < 16 * CIN; i += 256) {
        const int row = i >> 8, col = i & (CIN - 1);
        float v = 0.0f;
        if      (row == 0) v = w_reg[col];
        else if (row == 1) v = w_reg[CIN + col];
        else if (row == 2) v = w_conf[col];
        wpad[i] = v;
    }
    __syncthreads();

    const int lane = tid & 31;
    const int wave = tid >> 5;
    const int gblk = blockIdx.x;                     // 1024 blocks
    const int b    = gblk >> 6;                      // 64 blocks per batch
    const int pix0 = ((gblk & 63) << 8) + (wave << 5); // 256 px/block, 32 px/wave

    const int n    = lane & 15;                      // A: output row / B: pixel col
    const int koff = (lane >> 4) << 1;               // 0 or 2

    const float* __restrict__ xb   = x + (size_t)b * CIN * HWSZ;
    const float* __restrict__ wrow = &wpad[n * CIN + koff];

    v8f acc0 = {}, acc1 = {};

    #pragma unroll 4
    for (int k = 0; k < CIN; k += 4) {
        // A fragment: one ds_load_b64 (row-contiguous in padded LDS)
        v2f a = *reinterpret_cast<const v2f*>(wrow + k);

        const float* xk = xb + (size_t)(k + koff) * HWSZ + pix0 + n;
        v2f b0, b1;
        b0.x = xk[0];   b0.y = xk[HWSZ];             // tile 0, channels koff / koff+1
        b1.x = xk[16];  b1.y = xk[HWSZ + 16];        // tile 1

        acc0 = __builtin_amdgcn_wmma_f32_16x16x4_f32(
                   false, a, false, b0, (short)0, acc0, false, false);
        acc1 = __builtin_amdgcn_wmma_f32_16x16x4_f32(
                   false, a, false, b1, (short)0, acc1, false, false);
    }

    const float br0 = b_reg[0], br1 = b_reg[1], bc = b_conf[0];

    if (lane < 16) {
        // tile 0
        {
            float r0 = acc0[0] + br0;
            float r1 = acc0[1] + br1;
            float cf = 1.0f / (1.0f + __expf(-(acc0[2] + bc)));
            float sv = sqrtf(r0 * r0 + r1 * r1) * cf;
            const int pix = pix0 + lane;
            const size_t o = (size_t)b * HWSZ + pix;
            payload[o] = make_float4(r0, r1, cf, sv);
            scores[o]  = sv;
        }
        // tile 1
        {
            float r0 = acc1[0] + br0;
            float r1 = acc1[1] + br1;
            float cf = 1.0f / (1.0f + __expf(-(acc1[2] + bc)));
            float sv = sqrtf(r0 * r0 + r1 * r1) * cf;
            const int pix = pix0 + 16 + lane;
            const size_t o = (size_t)b * HWSZ + pix;
            payload[o] = make_float4(r0, r1, cf, sv);
            scores[o]  = sv;
        }
    }
}

// ---------------------------------------------------------------------------
// Kernel 2: per-batch exact top-300 via 4-pass radix select on float bits
// (scores are non-negative -> uint-monotonic), then bitonic sort of 512
// 64-bit keys ((bits<<32)|~idx : score desc, index asc), then gather+grid.
// One 512-thread block per batch.
// ---------------------------------------------------------------------------
__global__ __launch_bounds__(512) void topk_kernel(
    const float4* __restrict__ payload, const float* __restrict__ scores,
    float* __restrict__ out)
{
    __shared__ unsigned int       hist[256];
    __shared__ unsigned long long sel[512];
    __shared__ unsigned int       sh_prefix, sh_remaining, cntHi, cntEq;

    const int b   = blockIdx.x;
    const int tid = threadIdx.x;
    const float* __restrict__ sc = scores + (size_t)b * HWSZ;

    if (tid == 0) { sh_prefix = 0u; sh_remaining = KSEL; }
    __syncthreads();

    for (int pass = 0; pass < 4; ++pass) {
        const int shift = 24 - 8 * pass;
        const unsigned maskHi = (pass == 0) ? 0u : (0xFFFFFFFFu << (shift + 8));
        if (tid < 256) hist[tid] = 0u;
        __syncthreads();
        const unsigned pfx = sh_prefix;
        for (int i = tid; i < HWSZ; i += 512) {
            const unsigned bv = __float_as_uint(sc[i]);
            if ((bv & maskHi) == pfx)
                atomicAdd(&hist[(bv >> shift) & 0xFF], 1u);
        }
        __syncthreads();
        if (tid == 0) {
            unsigned rem = sh_remaining, cum = 0;
            int d = 255;
            for (; d >= 1; --d) {
                const unsigned c = hist[d];
                if (cum + c >= rem) break;
                cum += c;
            }
            sh_remaining = rem - cum;                 // equals still needed at this digit
            sh_prefix    = pfx | ((unsigned)d << shift);
        }
        __syncthreads();
    }

    const unsigned T   = sh_prefix;       // exact threshold bit pattern
    const unsigned rem = sh_remaining;    // #elements == T to take; (KSEL-rem) are > T
    if (tid == 0) { cntHi = 0u; cntEq = 0u; }
    sel[tid] = 0ull;                      // padding sorts last
    __syncthreads();

    for (int i = tid; i < HWSZ; i += 512) {
        const unsigned bv = __float_as_uint(sc[i]);
        if (bv > T) {
            const unsigned p = atomicAdd(&cntHi, 1u);
            if (p < KSEL)
                sel[p] = ((unsigned long long)bv << 32) | (unsigned)(~(unsigned)i);
        } else if (bv == T) {
            const unsigned q = atomicAdd(&cntEq, 1u);
            if (q < rem)   // equals fill slots [KSEL-rem, KSEL) from the top
                sel[(KSEL - 1) - q] =
                    ((unsigned long long)bv << 32) | (unsigned)(~(unsigned)i);
        }
    }
    __syncthreads();

    // bitonic sort, descending, 512 elements / 512 threads
    for (int k = 2; k <= 512; k <<= 1) {
        for (int j = k >> 1; j > 0; j >>= 1) {
            const int ixj = tid ^ j;
            if (ixj > tid) {
                const unsigned long long a = sel[tid], c = sel[ixj];
                const bool desc = ((tid & k) == 0);
                if (desc ? (a < c) : (a > c)) { sel[tid] = c; sel[ixj] = a; }
            }
            __syncthreads();
        }
    }

    if (tid < KSEL) {
        const unsigned long long key = sel[tid];
        const unsigned idx = ~(unsigned)(key & 0xFFFFFFFFull);
        const float4 p = payload[(size_t)b * HWSZ + idx];
        float* o = out + ((size_t)b * KSEL + tid) * 3;
        o[0] = (float)(idx >> 7)  + p.x;   // k // 128 + reg0
        o[1] = (float)(idx & 127) + p.y;   // k %  128 + reg1
        o[2] = p.z;                        // conf
    }
}

// ---------------------------------------------------------------------------
extern "C" void kernel_launch(void* const* d_in, const int* in_sizes, int n_in,
                              void* d_out, int out_size, void* d_ws, size_t ws_size,
                              hipStream_t stream)
{
    const float* x      = (const float*)d_in[0];
    const float* w_reg  = (const float*)d_in[1];
    const float* b_reg  = (const float*)d_in[2];
    const float* w_conf = (const float*)d_in[3];
    const float* b_conf = (const float*)d_in[4];
    float* out = (float*)d_out;

    float4* payload = (float4*)d_ws;                                   // 4 MiB
    float*  scores  = (float*)((char*)d_ws + (size_t)BATCH * HWSZ * sizeof(float4)); // 1 MiB

    head_gemm_kernel<<<1024, 256, 0, stream>>>(x, w_reg, b_reg, w_conf, b_conf,
                                               payload, scores);
    topk_kernel<<<BATCH, 512, 0, stream>>>(payload, scores, out);
}